// CPUForgetMult_42417097016197
// MI455X (gfx1250) — compile-verified
//
#include <hip/hip_runtime.h>
#include <hip/hip_bf16.h>
#include <stdint.h>

// h_t = f_t*x_t + (1-f_t)*h_{t-1}, SEQ=1024, B*H = 32768 independent channels.
// Streaming-bandwidth bound: ~403 MB -> ~17 us floor at 23.3 TB/s.
// One thread per channel; TDM (tensor_load_to_lds) ring-buffered pipeline
// to exceed the 63-entry LOADcnt in-flight cap. Compute reads staged tiles
// via ds_load_b32; results leave via non-temporal global stores.

#define SEQ_N  1024
#define BH_N   32768            // B*H = 32*1024
#define TT     16               // timesteps per TDM tile
#define DEPTH  4                // ring buffer depth (3 tiles in flight)
#define WPB    4                // waves per block
#define BLOCK  (WPB * 32)       // 128 threads
#define NTILE  (SEQ_N / TT)     // 64 tile steps

typedef unsigned int       u32;
typedef unsigned long long u64;
typedef u32 v4u __attribute__((ext_vector_type(4)));
typedef int v8i __attribute__((ext_vector_type(8)));
typedef int v4i __attribute__((ext_vector_type(4)));

// Issue one TDM load: 2D tile [TT rows x 32 channels] of f32 from a
// [SEQ_N x BH_N] row-major tensor into LDS at byte offset lds_off.
// D# group0: count=1 | lds_addr | global_addr(57b) | type=2.
// D# group1: data_size=4B, tensor_dim0=BH_N, tensor_dim1=SEQ_N,
//            tile_dim0=32, tile_dim1=TT, tensor_dim0_stride=BH_N.
__device__ __forceinline__ void tdm_load_tile(u32 lds_off, u64 gaddr) {
  v4u g0;
  g0.x = 1u;                                             // count=1, no gather
  g0.y = lds_off;                                        // lds_addr (bytes)
  g0.z = (u32)(gaddr & 0xFFFFFFFFu);                     // global_addr[31:0]
  g0.w = ((u32)(gaddr >> 32) & 0x01FFFFFFu) | (2u << 30);// addr[56:32] | type=2

  v8i g1;
  g1[0] = (int)(2u << 16);                               // data_size=2 (4 bytes)
  g1[1] = (int)(((u32)BH_N & 0xFFFFu) << 16);            // tensor_dim0 lo16
  g1[2] = (int)((((u32)BH_N >> 16) & 0xFFFFu)
              | (((u32)SEQ_N & 0xFFFFu) << 16));         // dim0 hi16 | dim1 lo16
  g1[3] = (int)((((u32)SEQ_N >> 16) & 0xFFFFu)
              | (32u << 16));                            // dim1 hi16 | tile_dim0=32
  g1[4] = (int)TT;                                       // tile_dim1=TT, tile_dim2=0
  g1[5] = (int)BH_N;                                     // tensor_dim0_stride lo32
  g1[6] = 0;                                             // stride0 hi16 | stride1 lo16
  g1[7] = 0;

  v4i z4 = {0, 0, 0, 0};
  v8i z8 = {0, 0, 0, 0, 0, 0, 0, 0};
  // clang-23 lane: 6-arg form (uint32x4, int32x8, int32x4, int32x4, int32x8, i32)
  __builtin_amdgcn_tensor_load_to_lds(g0, g1, z4, z4, z8, 0);
}

__global__ void __launch_bounds__(BLOCK)
forget_scan_tdm_kernel(const float* __restrict__ f,
                       const float* __restrict__ x,
                       const float* __restrict__ h0,
                       float* __restrict__ out) {
  // Per-wave private staging: [wave][depth][f|x][TT*32] floats = 64 KiB/block.
  __shared__ float smem[WPB][DEPTH][2][TT * 32];

  const int lane = (int)(threadIdx.x & 31u);
  const int wave = __builtin_amdgcn_readfirstlane((int)(threadIdx.x >> 5));
  const int chan_base = (int)blockIdx.x * BLOCK + wave * 32;
  const int chan = chan_base + lane;

  // Low 32 bits of a flat LDS pointer are the LDS byte address.
  // (This ptrtoint also makes smem "escape", so the asm memory clobbers
  //  below force the compiler to re-read LDS after each tensorcnt wait.)
  const u32 lds_base = (u32)(uintptr_t)(&smem[wave][0][0][0]);
  const u32 BUF_BYTES = (u32)(TT * 32 * 4);              // 2 KiB per tile buffer

  const u64 fbase = (u64)(uintptr_t)f + (u64)chan_base * 4u;
  const u64 xbase = (u64)(uintptr_t)x + (u64)chan_base * 4u;
  const u64 step  = (u64)TT * (u64)BH_N * 4u;            // bytes per tile in time

  // Prologue: prefetch DEPTH-1 tiles (each tile = one f op + one x op).
#pragma unroll
  for (int d = 0; d < DEPTH - 1; ++d) {
    tdm_load_tile(lds_base + (u32)(d * 2 + 0) * BUF_BYTES, fbase + (u64)d * step);
    tdm_load_tile(lds_base + (u32)(d * 2 + 1) * BUF_BYTES, xbase + (u64)d * step);
  }

  float h = h0[chan];

  for (int k = 0; k < NTILE; ++k) {
    // Prefetch tile k+3 into the slot last read at iteration k-1 (wave-private,
    // so plain program order is enough), then retire tile k (in-order).
    if (k + (DEPTH - 1) < NTILE) {
      const int d = (k + (DEPTH - 1)) & (DEPTH - 1);
      const u64 adv = (u64)(k + (DEPTH - 1)) * step;
      tdm_load_tile(lds_base + (u32)(d * 2 + 0) * BUF_BYTES, fbase + adv);
      tdm_load_tile(lds_base + (u32)(d * 2 + 1) * BUF_BYTES, xbase + adv);
      __builtin_amdgcn_s_wait_tensorcnt(6);  // 4 tiles out -> keep 3 newest
    } else if (k + 2 < NTILE) {
      __builtin_amdgcn_s_wait_tensorcnt(4);
    } else if (k + 1 < NTILE) {
      __builtin_amdgcn_s_wait_tensorcnt(2);
    } else {
      __builtin_amdgcn_s_wait_tensorcnt(0);
    }
    // Tile k is now resident in LDS; forbid hoisting/caching of the reads.
    asm volatile("" : : "s"(lds_base) : "memory");

    const int buf = k & (DEPTH - 1);
    int out_idx = k * TT * BH_N + chan;
#pragma unroll
    for (int t = 0; t < TT; ++t) {
      const float fv = smem[wave][buf][0][t * 32 + lane];  // ds_load_b32
      const float xv = smem[wave][buf][1][t * 32 + lane];  // ds_load_b32
      h = fmaf(fv, xv, (1.0f - fv) * h);     // f*x + (1-f)*h
      __builtin_nontemporal_store(h, out + out_idx);
      out_idx += BH_N;
    }
  }
}

extern "C" void kernel_launch(void* const* d_in, const int* in_sizes, int n_in,
                              void* d_out, int out_size, void* d_ws, size_t ws_size,
                              hipStream_t stream) {
  (void)in_sizes; (void)n_in; (void)out_size; (void)d_ws; (void)ws_size;
  const float* f  = (const float*)d_in[0];
  const float* x  = (const float*)d_in[1];
  const float* h0 = (const float*)d_in[2];
  float* out = (float*)d_out;

  forget_scan_tdm_kernel<<<dim3(BH_N / BLOCK), dim3(BLOCK), 0, stream>>>(f, x, h0, out);
}